// MultiCompartmentNeuron_35699768165144
// MI455X (gfx1250) — compile-verified
//
#include <hip/hip_runtime.h>

typedef float v2f __attribute__((ext_vector_type(2)));
typedef float v4f __attribute__((ext_vector_type(4)));
typedef float v8f __attribute__((ext_vector_type(8)));
typedef int   v4i __attribute__((ext_vector_type(4)));

typedef __attribute__((address_space(1))) v4i as1_v4i;
typedef __attribute__((address_space(3))) v4i as3_v4i;

#define ROWS 16        // neurons per wave tile
#define RSTRIDE 132    // padded LDS row stride in floats (bank-conflict-free A reads)
#define WAVES 4        // waves per block
#define TILEF (ROWS * RSTRIDE)

__launch_bounds__(128)
__global__ void MultiCompartmentNeuron_kernel(const float* __restrict__ syn,
                                              const float* __restrict__ soma_v,
                                              const float* __restrict__ dend_pot,
                                              const float* __restrict__ m_in,
                                              const float* __restrict__ h_in,
                                              const float* __restrict__ n_in,
                                              const float* __restrict__ ca_in,
                                              const float* __restrict__ adapt_in,
                                              const float* __restrict__ pre_in,
                                              const float* __restrict__ post_in,
                                              float* __restrict__ out,
                                              int N) {
  __shared__ float smem[WAVES * TILEF];
  const int lane = threadIdx.x & 31;
  const int wv   = threadIdx.x >> 5;
  const int tile = blockIdx.x * WAVES + wv;
  const int n0   = tile * ROWS;
  const bool active = (n0 < N);   // wave-uniform; N is a multiple of 16

  float* lds    = &smem[wv * TILEF];
  float* sdend  = lds;         // [0..127]  dend_in / dend_pot_new staging
  float* sstage = lds + 160;   // [160..367] packed output staging (16 x 13)

  if (active) {
    // ---------- Phase A: stage 16x128 f32 tile (contiguous 8 KB) into LDS ----------
    const float* gsrc = syn + (size_t)n0 * 128;
#if __has_builtin(__builtin_amdgcn_global_load_async_to_lds_b128)
#pragma unroll
    for (int r = 0; r < ROWS; ++r) {
      __builtin_amdgcn_global_load_async_to_lds_b128(
          (as1_v4i*)(gsrc + r * 128 + lane * 4),
          (as3_v4i*)(lds + r * RSTRIDE + lane * 4),
          0, 0);
    }
#if __has_builtin(__builtin_amdgcn_s_wait_asynccnt)
    __builtin_amdgcn_s_wait_asynccnt(0);
#else
    asm volatile("s_wait_asynccnt 0" ::: "memory");
#endif
#else
#pragma unroll
    for (int r = 0; r < ROWS; ++r) {
      v4f t = *(const v4f*)(gsrc + r * 128 + lane * 4);
      *(v4f*)(lds + r * RSTRIDE + lane * 4) = t;
    }
#endif

    // ---------- Phase B: WMMA segmented reduction -> dend_in[16 neurons][8 branches]
    // A (16x4 f32): lane L<16 holds (M=L, K=0,1); lane L>=16 holds (M=L-16, K=2,3).
    // B (4x16 f32): ones in column (j>>2) selects the branch; value is K-independent.
    const int mA = lane & 15;
    const int kh = lane >> 4;
    v8f acc = {0.f, 0.f, 0.f, 0.f, 0.f, 0.f, 0.f, 0.f};
#pragma unroll
    for (int j = 0; j < 32; ++j) {
      v2f a = *(const v2f*)(lds + mA * RSTRIDE + j * 4 + 2 * kh);
      float sel = ((lane & 15) == (j >> 2)) ? 1.0f : 0.0f;
      v2f b = {sel, sel};
      acc = __builtin_amdgcn_wmma_f32_16x16x4_f32(
          /*neg_a=*/false, a, /*neg_b=*/false, b,
          /*c_mod=*/(short)0, acc, /*reuse_a=*/false, /*reuse_b=*/false);
    }
    // D layout: lane L holds column n = L&15; c[r] is row m = 8*(L>>4)+r.
    {
      const int nD = lane & 15;
      if (nD < 8) {
#pragma unroll
        for (int r = 0; r < 8; ++r) sdend[(kh * 8 + r) * 8 + nD] = acc[r];
      }
    }
  }
  __syncthreads();

  // ---------- Phase C: per-(neuron,branch) NMDA / dendrite update ----------
  if (active) {
#pragma unroll
    for (int it = 0; it < 4; ++it) {
      int e  = it * 32 + lane;     // 0..127
      int mm = e >> 3;
      int b  = e & 7;
      float din  = sdend[e];
      float dp   = dend_pot[(size_t)n0 * 8 + e];            // coalesced
      float nmda = 1.0f / (1.0f + 0.28f * expf(-0.062f * dp));
      float dpn  = dp + (din * nmda - dp) * 0.01f;          // DT/TAU_DEND
      sdend[e] = dpn;
      sstage[mm * 13 + 3 + b] = dpn;
    }
  }
  __syncthreads();

  // ---------- Phase D: per-neuron HH soma update (16 lanes/wave) ----------
  if (active && lane < 16) {
    const int g = n0 + lane;
    v4f s0 = *(const v4f*)(sdend + lane * 8);
    v4f s1 = *(const v4f*)(sdend + lane * 8 + 4);
    float id = ((s0.x + s0.y) + (s0.z + s0.w)) + ((s1.x + s1.y) + (s1.z + s1.w));

    float v  = soma_v[g];
    float am = 0.1f * (v + 40.0f) / (1.0f - expf(-(v + 40.0f) * 0.1f));
    float bm = 4.0f * expf(-(v + 65.0f) * (1.0f / 18.0f));
    float ah = 0.07f * expf(-(v + 65.0f) * 0.05f);
    float bh = 1.0f / (1.0f + expf(-(v + 35.0f) * 0.1f));
    float an = 0.01f * (v + 55.0f) / (1.0f - expf(-(v + 55.0f) * 0.1f));
    float bn = 0.125f * expf(-(v + 65.0f) * 0.0125f);

    float m0 = m_in[g], h0 = h_in[g], q0 = n_in[g];
    float mN = fminf(fmaxf(m0 + (am * (1.0f - m0) - bm * m0) * 0.1f, 0.0f), 1.0f);
    float hN = fminf(fmaxf(h0 + (ah * (1.0f - h0) - bh * h0) * 0.1f, 0.0f), 1.0f);
    float qN = fminf(fmaxf(q0 + (an * (1.0f - q0) - bn * q0) * 0.1f, 0.0f), 1.0f);

    float i_na   = 180.0f * mN * mN * mN * hN * (v - 50.0f);   // GNA=120*1.5, ENA=50
    float i_k    = 36.0f * qN * qN * qN * qN * (v + 77.0f);    // EK=-77
    float i_leak = 10.0f * (v + 70.0f);                        // V_REST=-70
    float i_ad   = -(adapt_in[g] * expf(-0.001f));             // DT/ADAPT_TAU
    float v_new  = v + (id + i_na + i_k + i_leak + i_ad) * 0.001f;  // DT/CM

    float spike = (v_new > -55.0f) ? 1.0f : 0.0f;
    float v_out = (spike > 0.0f) ? -70.0f : v_new;
    float caN   = fmaxf((ca_in[g] + 0.1f * spike) * expf(-0.005f), 0.05f);
    float preN  = pre_in[g] * expf(-0.005f);
    float postN = post_in[g] * expf(-0.005f) + spike;

    sstage[lane * 13 + 0]  = spike;
    sstage[lane * 13 + 1]  = v_out;
    sstage[lane * 13 + 2]  = caN;
    sstage[lane * 13 + 11] = preN;
    sstage[lane * 13 + 12] = postN;
  }
  __syncthreads();

  // ---------- Phase E: coalesced packed-output writeback (16 x 13 floats) ----------
  if (active) {
    const size_t ob = (size_t)n0 * 13;
    for (int e = lane; e < ROWS * 13; e += 32) out[ob + e] = sstage[e];
  }
}

extern "C" void kernel_launch(void* const* d_in, const int* in_sizes, int n_in_bufs,
                              void* d_out, int out_size, void* d_ws, size_t ws_size,
                              hipStream_t stream) {
  (void)n_in_bufs; (void)out_size; (void)d_ws; (void)ws_size;
  const float* syn    = (const float*)d_in[0];
  const float* somav  = (const float*)d_in[1];
  const float* dendp  = (const float*)d_in[2];
  const float* m_in   = (const float*)d_in[3];
  const float* h_in   = (const float*)d_in[4];
  const float* nn_in  = (const float*)d_in[5];
  const float* ca_in  = (const float*)d_in[6];
  const float* ad_in  = (const float*)d_in[7];
  const float* pre_in = (const float*)d_in[8];
  const float* po_in  = (const float*)d_in[9];
  float* out = (float*)d_out;

  const int N = in_sizes[1];                 // soma_v has N elements
  const int tiles = (N + ROWS - 1) / ROWS;   // 16 neurons per wave
  const int blocks = (tiles + WAVES - 1) / WAVES;
  MultiCompartmentNeuron_kernel<<<blocks, WAVES * 32, 0, stream>>>(
      syn, somav, dendp, m_in, h_in, nn_in, ca_in, ad_in, pre_in, po_in, out, N);
}